// GraphAttentionLayer_68831145885835
// MI455X (gfx1250) — compile-verified
//
#include <hip/hip_runtime.h>

typedef __attribute__((ext_vector_type(2))) float v2f;
typedef __attribute__((ext_vector_type(8))) float v8f;

#define D 128
#define LEAKY 0.2f

// ---------- u1 = W_map @ w1, u2 = W_map @ w2 (128 threads, 1 block) ----------
__global__ void gat_compute_u(const float* __restrict__ Wmap,
                              const float* __restrict__ w1,
                              const float* __restrict__ w2,
                              float* __restrict__ u) {
    int k = threadIdx.x;              // 0..127
    float s1 = 0.f, s2 = 0.f;
    #pragma unroll 8
    for (int j = 0; j < D; ++j) {
        float wv = Wmap[k * D + j];
        s1 += wv * w1[j];
        s2 += wv * w2[j];
    }
    u[k] = s1;
    u[D + k] = s2;
}

// ---------- sa1[i] = x[i]·u1 + b1 ; sa2[i] = x[i]·u2 + b2 (wave per node) ----------
__global__ void gat_sa(const float* __restrict__ X, const float* __restrict__ u,
                       const float* __restrict__ b1, const float* __restrict__ b2,
                       float* __restrict__ sa1, float* __restrict__ sa2, int N) {
    int wave = threadIdx.x >> 5, lane = threadIdx.x & 31;
    int i = blockIdx.x * 8 + wave;
    if (i >= N) return;
    float s1 = 0.f, s2 = 0.f;
    #pragma unroll
    for (int t = 0; t < 4; ++t) {
        int k = lane + 32 * t;
        float xv = X[(long long)i * D + k];
        s1 += xv * u[k];
        s2 += xv * u[D + k];
    }
    #pragma unroll
    for (int off = 16; off > 0; off >>= 1) {
        s1 += __shfl_xor(s1, off, 32);
        s2 += __shfl_xor(s2, off, 32);
    }
    if (lane == 0) {
        sa1[i] = s1 + b1[0];
        sa2[i] = s2 + b2[0];
    }
}

// ---------- value = x @ kernel via V_WMMA_F32_16X16X4_F32 ----------
// One wave computes a 16x128 output strip: 8 accumulators (v8f each).
__global__ void gat_value_wmma(const float* __restrict__ X,
                               const float* __restrict__ Km,
                               float* __restrict__ V, int N, int nTiles) {
    int wave = threadIdx.x >> 5, lane = threadIdx.x & 31;
    int tile = blockIdx.x * 8 + wave;
    if (tile >= nTiles) return;              // wave-uniform: EXEC stays all-ones
    int row0 = tile * 16;
    int mrow = lane & 15;                    // M (A) / N (B,C,D) index within tile
    int khalf = lane >> 4;                   // 0 -> K0,K1 ; 1 -> K2,K3

    v8f acc[8];
    #pragma unroll
    for (int t = 0; t < 8; ++t) { v8f z = {}; acc[t] = z; }

    for (int k0 = 0; k0 < D; k0 += 4) {
        // A fragment: 16x4 f32. Lane L<16 holds (M=L, K=k0..k0+1); L>=16 holds K=k0+2..k0+3
        int ar = row0 + mrow; if (ar > N - 1) ar = N - 1;   // clamp keeps EXEC full
        const float* xr = X + (long long)ar * D + k0 + khalf * 2;
        v2f a; a.x = xr[0]; a.y = xr[1];
        #pragma unroll
        for (int t = 0; t < 8; ++t) {
            // B fragment: 4x16 f32 (K x N). Lane L<16 holds (N=L, K=k0..k0+1); L>=16 K=k0+2..+3
            int ncol = t * 16 + mrow;
            int kr = k0 + khalf * 2;
            v2f b; b.x = Km[kr * D + ncol]; b.y = Km[(kr + 1) * D + ncol];
            acc[t] = __builtin_amdgcn_wmma_f32_16x16x4_f32(
                false, a, false, b, (short)0, acc[t], false, false);
        }
    }
    // D layout: VGPR v, lanes 0-15 -> M=v ; lanes 16-31 -> M=v+8 ; N = lane&15
    int rbase = row0 + (lane >> 4) * 8;
    #pragma unroll
    for (int t = 0; t < 8; ++t) {
        int n = t * 16 + mrow;
        #pragma unroll
        for (int v = 0; v < 8; ++v) {
            int r = rbase + v;
            if (r < N) V[(long long)r * D + n] = acc[t][v];
        }
    }
}

// ---------- CSR build step 1: zero degree + cursor ----------
__global__ void gat_zero(int* __restrict__ deg, int* __restrict__ cursor, int N) {
    int i = blockIdx.x * blockDim.x + threadIdx.x;
    if (i < N) { deg[i] = 0; cursor[i] = 0; }
}

// ---------- CSR build step 2: degree histogram ----------
__global__ void gat_count(const int* __restrict__ er, int* __restrict__ deg, int E) {
    int e = blockIdx.x * blockDim.x + threadIdx.x;
    if (e < E) atomicAdd(&deg[er[e]], 1);
}

// ---------- CSR build step 3: exclusive scan -> rowptr (single block, 1024 thr) ----------
__global__ void gat_scan(const int* __restrict__ deg, int* __restrict__ rowptr, int N) {
    __shared__ int sums[1024];
    int tid = threadIdx.x;
    int chunk = (N + 1023) / 1024;
    int start = tid * chunk;
    int end = start + chunk; if (end > N) end = N; if (start > N) start = N;
    int s = 0;
    for (int i = start; i < end; ++i) s += deg[i];
    sums[tid] = s;
    __syncthreads();
    // Hillis-Steele inclusive scan over per-thread sums
    for (int off = 1; off < 1024; off <<= 1) {
        int v = 0;
        if (tid >= off) v = sums[tid - off];
        __syncthreads();
        sums[tid] += v;
        __syncthreads();
    }
    int run = (tid > 0) ? sums[tid - 1] : 0;   // exclusive base for this chunk
    for (int i = start; i < end; ++i) { rowptr[i] = run; run += deg[i]; }
    if (tid == 0) rowptr[N] = sums[1023];
}

// ---------- CSR build step 4: scatter (score computed inline, leaky relu) ----------
__global__ void gat_scatter(const float* __restrict__ adj, const int* __restrict__ er,
                            const int* __restrict__ ec, const float* __restrict__ sa1,
                            const float* __restrict__ sa2, const int* __restrict__ rowptr,
                            int* __restrict__ cursor, int* __restrict__ s_col,
                            float* __restrict__ s_score, int E) {
    int e = blockIdx.x * blockDim.x + threadIdx.x;
    if (e >= E) return;
    int r = er[e], c = ec[e];
    float a = adj[e];
    float v = a * sa1[r] + a * sa2[c];
    v = (v >= 0.f) ? v : LEAKY * v;
    int pos = rowptr[r] + atomicAdd(&cursor[r], 1);
    s_score[pos] = v;
    s_col[pos]   = c;
}

// ---------- row kernel: softmax over segment + weighted gather, NO float atomics ----------
// One wave per row; lane owns output dims [4*lane, 4*lane+4) as float4.
__global__ void gat_row(const int* __restrict__ rowptr, const int* __restrict__ s_col,
                        const float* __restrict__ s_score, const float* __restrict__ V,
                        const float* __restrict__ bias, float* __restrict__ out, int N) {
    int wave = threadIdx.x >> 5, lane = threadIdx.x & 31;
    int r = blockIdx.x * 8 + wave;
    if (r >= N) return;
    int beg = rowptr[r], end = rowptr[r + 1];

    // segment max (order-independent)
    float m = -3.402823466e+38f;
    for (int j = beg + lane; j < end; j += 32) m = fmaxf(m, s_score[j]);
    #pragma unroll
    for (int off = 16; off > 0; off >>= 1) m = fmaxf(m, __shfl_xor(m, off, 32));

    // segment sum of exp
    float s = 0.f;
    for (int j = beg + lane; j < end; j += 32) s += __expf(s_score[j] - m);
    #pragma unroll
    for (int off = 16; off > 0; off >>= 1) s += __shfl_xor(s, off, 32);
    float inv = (end > beg) ? 1.f / s : 0.f;

    // weighted gather-accumulate: one b128 load per lane per edge (coalesced 512B/row/edge)
    float4 acc = make_float4(0.f, 0.f, 0.f, 0.f);
    for (int j = beg; j < end; ++j) {
        float w = __expf(s_score[j] - m) * inv;   // wave-uniform
        int c = s_col[j];
        float4 vc = ((const float4*)(V + (long long)c * D))[lane];
        acc.x += w * vc.x; acc.y += w * vc.y;
        acc.z += w * vc.z; acc.w += w * vc.w;
    }
    float4 b4 = ((const float4*)(bias + (long long)r * D))[lane];
    float4 o;
    o.x = acc.x + b4.x; o.y = acc.y + b4.y;
    o.z = acc.z + b4.z; o.w = acc.w + b4.w;
    ((float4*)(out + (long long)r * D))[lane] = o;
}

extern "C" void kernel_launch(void* const* d_in, const int* in_sizes, int n_in,
                              void* d_out, int out_size, void* d_ws, size_t ws_size,
                              hipStream_t stream) {
    const float* x        = (const float*)d_in[0];
    const float* adj_vals = (const float*)d_in[1];
    const int*   edge_row = (const int*)d_in[2];
    const int*   edge_col = (const int*)d_in[3];
    const float* W_map    = (const float*)d_in[4];
    const float* w1       = (const float*)d_in[5];
    const float* b1       = (const float*)d_in[6];
    const float* w2       = (const float*)d_in[7];
    const float* b2       = (const float*)d_in[8];
    const float* kern     = (const float*)d_in[9];
    const float* bias     = (const float*)d_in[10];
    float* out = (float*)d_out;

    const int N = in_sizes[0] / D;
    const int E = in_sizes[1];
    const long long Np = ((long long)N + 3) & ~3LL;   // keep 16B alignment of carve-outs

    // workspace carve-up (4-byte words, all offsets multiple of 4 words = 16B)
    float* ws = (float*)d_ws;
    float* u      = ws;                                  // 2*D
    float* sa1    = u + 2 * D;                           // Np
    float* sa2    = sa1 + Np;                            // Np
    float* value  = sa2 + Np;                            // N*D (D multiple of 4)
    int*   deg    = (int*)(value + (long long)N * D);    // Np
    int*   cursor = deg + Np;                            // Np
    int*   rowptr = cursor + Np;                         // Np+4
    int*   s_col  = rowptr + Np + 4;                     // E
    float* s_score= (float*)(s_col + (((long long)E + 3) & ~3LL)); // E

    // 1) u1/u2
    gat_compute_u<<<1, D, 0, stream>>>(W_map, w1, w2, u);
    // 2) per-node attention scalars
    gat_sa<<<(N + 7) / 8, 256, 0, stream>>>(x, u, b1, b2, sa1, sa2, N);
    // 3) value = x @ kernel with WMMA
    int nTiles = (N + 15) / 16;
    gat_value_wmma<<<(nTiles + 7) / 8, 256, 0, stream>>>(x, kern, value, N, nTiles);
    // 4) CSR build
    gat_zero<<<(N + 255) / 256, 256, 0, stream>>>(deg, cursor, N);
    gat_count<<<(E + 255) / 256, 256, 0, stream>>>(edge_row, deg, E);
    gat_scan<<<1, 1024, 0, stream>>>(deg, rowptr, N);
    gat_scatter<<<(E + 255) / 256, 256, 0, stream>>>(adj_vals, edge_row, edge_col,
                                                     sa1, sa2, rowptr, cursor,
                                                     s_col, s_score, E);
    // 5) fused softmax + SpMM per row (no float atomics anywhere)
    gat_row<<<(N + 7) / 8, 256, 0, stream>>>(rowptr, s_col, s_score, value, bias, out, N);
}